// ComplexLayerScale_25125558681903
// MI455X (gfx1250) — compile-verified
//
#include <hip/hip_runtime.h>

// ComplexLayerScale: out[b,t,d] = complex(x_r,x_i)[b,t,d] * complex(g_r,g_i)[d]
// Memory-bound streaming kernel: ~268 MB moved, ~0.1 GFLOP -> HBM roofline
// (~11.5 us at 23.3 TB/s). Strategy:
//   * gamma (4 KB) staged to LDS via CDNA5 async global->LDS copies
//     (global_load_async_to_lds_b128, ASYNCcnt-tracked, no VGPR roundtrip)
//   * B128 non-temporal global loads/stores for the 268 MB stream
//   * ds_load_b128 reads of gamma, pure-FMA complex multiply

typedef __attribute__((ext_vector_type(4))) float f32x4;

#define D_DIM 512          // channel dim from the reference (power of two)
#define BLOCK 256          // 8 wave32 per workgroup

__global__ __launch_bounds__(BLOCK) void complex_layer_scale_kernel(
    const float* __restrict__ xr,
    const float* __restrict__ xi,
    const float* __restrict__ gr,
    const float* __restrict__ gi,
    float* __restrict__ out,       // interleaved (re,im) pairs: complex64 layout
    long long n_chunks)            // number of 4-element chunks = (B*T*D)/4
{
    // s_g[0..511] = gamma_real, s_g[512..1023] = gamma_imag
    __shared__ __align__(16) float s_g[2 * D_DIM];

    const int tid = threadIdx.x;

    // ---- Stage gamma into LDS with CDNA5 async copies -------------------
    // 256 threads x 16B = 4 KB. Threads 0..127 -> gamma_real,
    // threads 128..255 -> gamma_imag. VDST = LDS byte offset (32-bit),
    // VADDR = 64-bit global address (GV mode, saddr = off).
    {
        const float* gsrc = (tid < 128) ? (gr + tid * 4)
                                        : (gi + (tid - 128) * 4);
        unsigned lds_off = (unsigned)(unsigned long long)(&s_g[tid * 4]);
        asm volatile("global_load_async_to_lds_b128 %0, %1, off"
                     :: "v"(lds_off), "v"(gsrc)
                     : "memory");
        // Each wave waits for its own async copies to land in LDS,
        // then the workgroup barrier makes all 4 KB visible to every wave.
        asm volatile("s_wait_asynccnt 0" ::: "memory");
    }
    __syncthreads();

    const long long c = (long long)blockIdx.x * BLOCK + tid;
    if (c >= n_chunks) return;

    const long long base = c << 2;                 // flat element index, 16B aligned
    // Streaming inputs: non-temporal B128 loads (no reuse, don't pollute L2).
    const f32x4 xrv = __builtin_nontemporal_load((const f32x4*)(xr + base));
    const f32x4 xiv = __builtin_nontemporal_load((const f32x4*)(xi + base));

    // Chunk of 4 stays within one D-row (D=512, base multiple of 4).
    const int d = (int)(base & (D_DIM - 1));       // 16B-aligned LDS offset
    const f32x4 grv = *(const f32x4*)(&s_g[d]);            // ds_load_b128
    const f32x4 giv = *(const f32x4*)(&s_g[D_DIM + d]);    // ds_load_b128

    // Complex multiply, interleave (re,im) pairs for complex64 output layout.
    f32x4 o0, o1;
    o0.x = fmaf(xrv.x, grv.x, -(xiv.x * giv.x));   // re0
    o0.y = fmaf(xrv.x, giv.x,  (xiv.x * grv.x));   // im0
    o0.z = fmaf(xrv.y, grv.y, -(xiv.y * giv.y));   // re1
    o0.w = fmaf(xrv.y, giv.y,  (xiv.y * grv.y));   // im1
    o1.x = fmaf(xrv.z, grv.z, -(xiv.z * giv.z));   // re2
    o1.y = fmaf(xrv.z, giv.z,  (xiv.z * grv.z));   // im2
    o1.z = fmaf(xrv.w, grv.w, -(xiv.w * giv.w));   // re3
    o1.w = fmaf(xrv.w, giv.w,  (xiv.w * grv.w));   // im3

    // Streaming output: non-temporal B128 stores (write-once, full cachelines).
    float* o = out + (base << 1);                  // 32B aligned
    __builtin_nontemporal_store(o0, (f32x4*)(o));
    __builtin_nontemporal_store(o1, (f32x4*)(o + 4));
}

extern "C" void kernel_launch(void* const* d_in, const int* in_sizes, int n_in,
                              void* d_out, int out_size, void* d_ws, size_t ws_size,
                              hipStream_t stream) {
    const float* xr = (const float*)d_in[0];   // x_real  [B,T,D] f32
    const float* xi = (const float*)d_in[1];   // x_imag  [B,T,D] f32
    const float* gr = (const float*)d_in[2];   // gamma_real [D] f32
    const float* gi = (const float*)d_in[3];   // gamma_imag [D] f32
    float* out = (float*)d_out;                // complex64 -> interleaved f32 pairs

    const long long n = (long long)in_sizes[0];        // B*T*D = 16,777,216
    const long long n_chunks = n >> 2;                 // 4 elements per thread
    const long long grid = (n_chunks + BLOCK - 1) / BLOCK;

    hipLaunchKernelGGL(complex_layer_scale_kernel,
                       dim3((unsigned)grid), dim3(BLOCK), 0, stream,
                       xr, xi, gr, gi, out, n_chunks);
}